// cmaBlock_31988916421178
// MI455X (gfx1250) — compile-verified
//
#include <hip/hip_runtime.h>
#include <hip/hip_bf16.h>

typedef __attribute__((ext_vector_type(2))) float v2f;
typedef __attribute__((ext_vector_type(8))) float v8f;

#define B_   256
#define C_   256
#define CH_  128
#define N_   196
#define NV4_ 49      // 196 floats = 49 float4
#define EPS_ 1e-5f

// ---------------------------------------------------------------------------
// Kernel 1: ysum[b*C+c] = sum over 196 spatial elements of y[b,c,:,:]
// One wave32 per row; lane-strided loads (coalesced), shuffle reduction.
// ---------------------------------------------------------------------------
__global__ __launch_bounds__(256) void ysum_kernel(const float* __restrict__ y,
                                                   float* __restrict__ ysum) {
    const int wave = threadIdx.x >> 5;
    const int lane = threadIdx.x & 31;
    const int row  = blockIdx.x * 8 + wave;      // b*C + c, 65536 rows
    const float* p = y + (size_t)row * N_;
    float s = 0.0f;
    #pragma unroll
    for (int i = lane; i < N_; i += 32) s += p[i];
    #pragma unroll
    for (int off = 16; off > 0; off >>= 1) s += __shfl_xor(s, off, 32);
    if (lane == 0) ysum[row] = s;
}

// ---------------------------------------------------------------------------
// Kernel 2: Wcomb = Wz @ Wv   ([256,128] x [128,256] -> [256,256], fp32 WMMA)
// One wave per 16x16 output tile; K-loop of V_WMMA_F32_16X16X4_F32.
// A 16x4 fragment: lanes 0-15 -> M=lane, K={k0,k0+1}; lanes 16-31 -> K={k0+2,k0+3}.
// C/D: vgpr v, lane -> D[M = v + 8*(lane>>4)][N = lane&15].
// ---------------------------------------------------------------------------
__global__ __launch_bounds__(256) void wcomb_kernel(const float* __restrict__ Wz,
                                                    const float* __restrict__ Wv,
                                                    float* __restrict__ Wcomb) {
    const int wave = threadIdx.x >> 5;
    const int lane = threadIdx.x & 31;
    const int tile = blockIdx.x * 8 + wave;       // 16x16 tiles over [256,256]
    const int tm   = (tile >> 4) << 4;
    const int tn   = (tile & 15) << 4;
    const int half = lane >> 4;
    const int r    = lane & 15;
    v8f acc = {};
    for (int k0 = 0; k0 < CH_; k0 += 4) {
        const int kb = k0 + half * 2;
        v2f a, b;
        const float* ap = Wz + (tm + r) * CH_ + kb;   // A[row][kb], A[row][kb+1]
        a.x = ap[0];
        a.y = ap[1];
        b.x = Wv[kb * C_ + tn + r];                   // B[kb][n]
        b.y = Wv[(kb + 1) * C_ + tn + r];             // B[kb+1][n]
        acc = __builtin_amdgcn_wmma_f32_16x16x4_f32(
            /*neg_a=*/false, a, /*neg_b=*/false, b,
            /*c_mod=*/(short)0, acc, /*reuse_a=*/false, /*reuse_b=*/false);
    }
    #pragma unroll
    for (int v = 0; v < 8; ++v)
        Wcomb[(tm + v + half * 8) * C_ + tn + r] = acc[v];
}

// ---------------------------------------------------------------------------
// Kernel 3: Zmat[b][c] = sum_k Ysum[b][k] * Wcomb[c][k]   (Ysum @ Wcomb^T)
// B-operand fragment B[k][n] = Wcomb[n*256 + k] -> consecutive float2 per lane.
// ---------------------------------------------------------------------------
__global__ __launch_bounds__(256) void zmat_kernel(const float* __restrict__ Ysum,
                                                   const float* __restrict__ Wcomb,
                                                   float* __restrict__ Zmat) {
    const int wave = threadIdx.x >> 5;
    const int lane = threadIdx.x & 31;
    const int tile = blockIdx.x * 8 + wave;
    const int tm   = (tile >> 4) << 4;   // over batch
    const int tn   = (tile & 15) << 4;   // over channels
    const int half = lane >> 4;
    const int r    = lane & 15;
    v8f acc = {};
    for (int k0 = 0; k0 < C_; k0 += 4) {
        const int kb = k0 + half * 2;
        v2f a, b;
        const float* ap = Ysum  + (tm + r) * C_ + kb;
        const float* bp = Wcomb + (tn + r) * C_ + kb; // transposed access
        a.x = ap[0]; a.y = ap[1];
        b.x = bp[0]; b.y = bp[1];
        acc = __builtin_amdgcn_wmma_f32_16x16x4_f32(
            false, a, false, b, (short)0, acc, false, false);
    }
    #pragma unroll
    for (int v = 0; v < 8; ++v)
        Zmat[(tm + v + half * 8) * C_ + tn + r] = acc[v];
}

// ---------------------------------------------------------------------------
// Kernel 4: out[b,c,:,:] = x[b,c,:,:] + (Zmat[b][c]-mu_c)*rsqrt(var_c+eps)*g_c + b_c
// float4 vectorized; 4 rows (b,c pairs) per 256-thread block.
// ---------------------------------------------------------------------------
__global__ __launch_bounds__(256) void out_kernel(const float* __restrict__ x,
                                                  const float* __restrict__ zmat,
                                                  const float* __restrict__ bw,
                                                  const float* __restrict__ bb,
                                                  const float* __restrict__ bm,
                                                  const float* __restrict__ bv,
                                                  float* __restrict__ out) {
    const int grp  = threadIdx.x >> 6;            // 0..3
    const int lane = threadIdx.x & 63;
    const int row  = blockIdx.x * 4 + grp;        // b*C + c
    const int c    = row & (C_ - 1);
    const float inv = rsqrtf(bv[c] + EPS_);
    const float val = (zmat[row] - bm[c]) * (inv * bw[c]) + bb[c];
    if (lane < NV4_) {
        const float4* xp = (const float4*)x + (size_t)row * NV4_ + lane;
        float4* op = (float4*)out + (size_t)row * NV4_ + lane;
        float4 v = *xp;
        v.x += val; v.y += val; v.z += val; v.w += val;
        *op = v;
    }
}

extern "C" void kernel_launch(void* const* d_in, const int* in_sizes, int n_in,
                              void* d_out, int out_size, void* d_ws, size_t ws_size,
                              hipStream_t stream) {
    // setup_inputs order: x, y, Wq, Wk, Wv, Wz, bn_weight, bn_bias, bn_mean, bn_var
    const float* x   = (const float*)d_in[0];
    const float* y   = (const float*)d_in[1];
    // d_in[2]=Wq, d_in[3]=Wk: mathematically unused (softmax over singleton axis == ones)
    const float* Wv  = (const float*)d_in[4];
    const float* Wz  = (const float*)d_in[5];
    const float* bw  = (const float*)d_in[6];
    const float* bb  = (const float*)d_in[7];
    const float* bm  = (const float*)d_in[8];
    const float* bv  = (const float*)d_in[9];
    float* out = (float*)d_out;

    float* ws    = (float*)d_ws;
    float* ysum  = ws;                 // 65536 floats
    float* wcomb = ws + 65536;         // 65536 floats
    float* zm    = ws + 131072;        // 65536 floats

    ysum_kernel <<<(B_ * C_) / 8, 256, 0, stream>>>(y, ysum);
    wcomb_kernel<<<(16 * 16) / 8,  256, 0, stream>>>(Wz, Wv, wcomb);
    zmat_kernel <<<(16 * 16) / 8,  256, 0, stream>>>(ysum, wcomb, zm);
    out_kernel  <<<(B_ * C_) / 4, 256, 0, stream>>>(x, zm, bw, bb, bm, bv, out);
}